// FullSpaceMapperFEATClusterLin_Net_71116068487224
// MI455X (gfx1250) — compile-verified
//
#include <hip/hip_runtime.h>
#include <math.h>

// ---------------- constants ----------------
#define NPIX      32768      // B * 128 * 128
#define OUT_DELTA 0          // (2,18,512) = 18432 floats
#define OUT_BLUR  18432      // (2,1,128,128) = 32768 floats
#define OUT_LD    51200      // loss_delta
#define OUT_LR    51201      // loss_reg
#define OUT_LTV   51202      // loss_tv

// workspace layout (float offsets)
#define WS_V      0          // folded conv weights, 5088 floats
#define WS_KC     5088       // folded bias constant (1)
#define WS_SUMS   5092       // 20
#define WS_CNTS   5112       // 20
#define WS_MEAN   5132       // 20
#define WS_LOGIT  5248       // 32768 (reused as FINAL after sigmoid stage)
#define WS_FINAL  5248
#define WS_EACH   38016      // 32768
#define WS_ASSIGN 70784      // 32768 (int)
#define WS_ZERO_N 38016      // zero [0, WS_LOGIT+32768) == [0,38016)

typedef __attribute__((ext_vector_type(2))) float v2f;
typedef __attribute__((ext_vector_type(8))) float v8f;

// per-attention-layer tables (l=0 -> fmap_25, l=1..17 -> fmaps[LAYER_NUM[l-1]])
__constant__ int c_cin[18]  = {512,512,512,512,512,512,512,512,256,256,128,128,64,64,32,32,16,16};
__constant__ int c_H[18]    = {4,4,8,16,32,32,64,64,256,256,256,256,256,256,256,256,256,256};
__constant__ int c_voff[18] = {0,512,1024,1536,2048,2560,3072,3584,4096,4352,4608,4736,4864,4928,4992,5024,5056,5072};
// 44 (layer, channel-chunk<=128) work items for the logits kernel
__constant__ int c_ck_l[44] = {0,0,0,0, 1,1,1,1, 2,2,2,2, 3,3,3,3, 4,4,4,4, 5,5,5,5, 6,6,6,6, 7,7,7,7,
                               8,8, 9,9, 10, 11, 12, 13, 14, 15, 16, 17};
__constant__ int c_ck_c0[44]= {0,128,256,384, 0,128,256,384, 0,128,256,384, 0,128,256,384,
                               0,128,256,384, 0,128,256,384, 0,128,256,384, 0,128,256,384,
                               0,128, 0,128, 0, 0, 0, 0, 0, 0, 0, 0};

struct FmapPtrs { const float* p[18]; };

// ---------------- K0: zero ----------------
__global__ void zero_kernel(float* ptr, int n) {
  int i = blockIdx.x * 256 + threadIdx.x;
  if (i < n) ptr[i] = 0.0f;
}

// ---------------- K1: mapper (rmsnorm + 3 matvec layers) ----------------
__global__ void mapper_kernel(const float* __restrict__ x, const float* __restrict__ mw,
                              const float* __restrict__ mb, float* __restrict__ out,
                              float* __restrict__ loss_delta) {
  __shared__ float h0[512], h1[512], red[256];
  const int blk = blockIdx.x, b = blk / 7, c = blk % 7, t = threadIdx.x;
  const float* xr = x + ((size_t)b * 18 + c) * 512;
  float a0 = xr[t], a1 = xr[t + 256];
  red[t] = a0 * a0 + a1 * a1;
  __syncthreads();
  for (int o = 128; o > 0; o >>= 1) { if (t < o) red[t] += red[t + o]; __syncthreads(); }
  float rs = rsqrtf(red[0] * (1.0f / 512.0f) + 1e-8f);
  h0[t] = a0 * rs; h0[t + 256] = a1 * rs;
  __syncthreads();
  float* hc = h0; float* hn = h1;
  const float wscale = (1.0f / 22.6274169979695f) * 0.1f;  // 1/sqrt(512)*LR_MUL
  for (int layer = 0; layer < 3; ++layer) {
    const float* wbase = mw + ((size_t)c * 3 + layer) * 512 * 512;
    const float* bbase = mb + ((size_t)c * 3 + layer) * 512;
    float vout[2];
    #pragma unroll
    for (int oo = 0; oo < 2; ++oo) {
      const int o = t + oo * 256;
      const float* wr = wbase + (size_t)o * 512;
      float acc = 0.0f;
      for (int k = 0; k < 512; ++k) acc = fmaf(hc[k], wr[k], acc);
      float v = acc * wscale + bbase[o] * 0.1f;
      v = (v >= 0.0f ? v : 0.2f * v) * 1.41421356237f;
      vout[oo] = v;
    }
    __syncthreads();                 // all reads of hc done
    hn[t] = vout[0]; hn[t + 256] = vout[1];
    __syncthreads();
    float* tmp = hc; hc = hn; hn = tmp;
  }
  float o1 = hc[t], o2 = hc[t + 256];
  out[((size_t)b * 18 + c) * 512 + t]       = o1;
  out[((size_t)b * 18 + c) * 512 + t + 256] = o2;
  red[t] = o1 * o1 + o2 * o2;
  __syncthreads();
  for (int o = 128; o > 0; o >>= 1) { if (t < o) red[t] += red[t + o]; __syncthreads(); }
  if (t == 0) atomicAdd(loss_delta, sqrtf(red[0]) * (1.0f / 14.0f));
}

// ---------------- K2: fold alw into conv1x1 weights ----------------
__global__ void fold_kernel(const float* __restrict__ afw, const float* __restrict__ afb,
                            const float* __restrict__ aw, const float* __restrict__ ab,
                            const float* __restrict__ alw, float* __restrict__ v,
                            float* __restrict__ KC) {
  const int l = blockIdx.x, t = threadIdx.x, cin = c_cin[l];
  const float* W  = (l == 0) ? afw : aw + (size_t)(l - 1) * 32 * 512;
  const float* Bv = (l == 0) ? afb : ab + (l - 1) * 32;
  const float inv = rsqrtf((float)cin);
  for (int i = t; i < cin; i += 256) {
    float s = 0.0f;
    for (int o = 0; o < 32; ++o) s = fmaf(alw[32 * l + o], W[(size_t)o * 512 + i], s);
    v[c_voff[l] + i] = s * inv;
  }
  if (t == 0) {
    float s = 0.0f;
    for (int o = 0; o < 32; ++o) s = fmaf(alw[32 * l + o], Bv[o], s);
    atomicAdd(KC, s);
  }
}

// ---------------- K3: folded conv -> per-pixel logit partials ----------------
__global__ void logits_partial_kernel(FmapPtrs fm, const float* __restrict__ v,
                                      float* __restrict__ logit_acc) {
  __shared__ float sv[128];
  const int chunk = blockIdx.y;
  const int l = c_ck_l[chunk], c0 = c_ck_c0[chunk];
  const int cin = c_cin[l], H = c_H[l];
  const int len = min(128, cin - c0);
  const int t = threadIdx.x;
  if (t < len) sv[t] = v[c_voff[l] + c0 + t];
  __syncthreads();
  const int p = blockIdx.x * 256 + t;
  const int b = p >> 14, q = p & 16383, y = q >> 7, x = q & 127;
  const int sy = (y * H) >> 7, sx = (x * H) >> 7;   // == arange(128)*H//128 sampling
  const size_t stride = (size_t)H * H;
  const float* base = fm.p[l] + ((size_t)b * cin + c0) * stride + (size_t)sy * H + sx;
  float acc = 0.0f;
  for (int i = 0; i < len; ++i) acc = fmaf(base[(size_t)i * stride], sv[i], acc);
  atomicAdd(&logit_acc[p], acc);
}

// ---------------- K3b: sigmoid ----------------
__global__ void sigmoid_kernel(const float* __restrict__ logit, const float* __restrict__ KC,
                               const float* __restrict__ alb, float* __restrict__ each) {
  const int p = blockIdx.x * 256 + threadIdx.x;
  const float lg = (logit[p] + KC[0]) * (1.0f / 24.0f) + alb[0];  // 1/sqrt(32*18)
  each[p] = 1.0f / (1.0f + expf(-lg));
}

// ---------------- K4: cluster assignment via f32 WMMA GEMM ----------------
// dist'(p,k) = -2*dot(blend_p, S_k[0:512]) + ||S_k||^2 - 2*(xp*sum(S_k[512:544]) + yp*sum(S_k[544:576]))
// (per-pixel ||f||^2 dropped: constant under argmin)
// N is padded 10 -> 16 with explicit zero rows in LDS so the B-fragment load is
// unconditional: no EXEC save/restore in the WMMA inner loop.
__global__ void cluster_assign_kernel(const float* __restrict__ blend,
                                      const float* __restrict__ S,
                                      int* __restrict__ assign) {
  __shared__ float sS[16 * 576];          // rows 10..15 zero-padded
  __shared__ float sSsq[10], sSx[10], sSy[10];
  __shared__ float tile[8][256];
  const int t = threadIdx.x;
  for (int i = t; i < 16 * 576; i += 256) sS[i] = (i < 10 * 576) ? S[i] : 0.0f;
  __syncthreads();
  if (t < 10) {
    const float* row = sS + t * 576;
    float q = 0.f, gx = 0.f, gy = 0.f;
    for (int j = 0; j < 576; ++j) q = fmaf(row[j], row[j], q);
    for (int j = 512; j < 544; ++j) gx += row[j];
    for (int j = 544; j < 576; ++j) gy += row[j];
    sSsq[t] = q; sSx[t] = gx; sSy[t] = gy;
  }
  __syncthreads();
  const int wave = t >> 5, lane = t & 31;
  const int tilei = blockIdx.x * 8 + wave;        // 2048 tiles of 16 pixels
  const int p0 = tilei * 16;
  const int b = p0 >> 14, q0 = p0 & 16383;
  const float* bb = blend + (size_t)b * 512 * 16384 + q0;
  const int m  = lane & 15;                       // pixel (A) / cluster (B) within half-wave
  const int hi = lane >> 4;                       // 0: K0/K1, 1: K2/K3 (ISA 16x4 f32 A layout)
  const float* srow = sS + m * 576;               // B[k][n] = S[n][k], padded rows are zero
  v8f acc = {0.f, 0.f, 0.f, 0.f, 0.f, 0.f, 0.f, 0.f};
  for (int k0 = 0; k0 < 512; k0 += 4) {
    const int ka = k0 + hi * 2;
    v2f a, bf;
    a.x  = bb[(size_t)(ka + 0) * 16384 + m];
    a.y  = bb[(size_t)(ka + 1) * 16384 + m];
    bf.x = srow[ka + 0];
    bf.y = srow[ka + 1];
    acc = __builtin_amdgcn_wmma_f32_16x16x4_f32(false, a, false, bf,
                                                (short)0, acc, false, false);
  }
  // D layout: VGPR r -> lanes0-15 M=r, lanes16-31 M=r+8, N = lane&15
  float* tw = tile[wave];
  #pragma unroll
  for (int r = 0; r < 8; ++r) tw[(r + hi * 8) * 16 + m] = acc[r];
  __syncthreads();
  if (lane < 16) {
    const int q = q0 + lane;
    const int y = q >> 7, x = q & 127;
    const float xp = x * (2.0f / 127.0f) - 1.0f;
    const float yp = y * (2.0f / 127.0f) - 1.0f;
    float best = 3.4e38f; int bi = 0;
    for (int n = 0; n < 10; ++n) {
      float d = -2.0f * tw[lane * 16 + n] + sSsq[n] - 2.0f * (xp * sSx[n] + yp * sSy[n]);
      if (d < best) { best = d; bi = n; }          // first-min tie-break like argmin
    }
    assign[p0 + lane] = b * 10 + bi;               // store segment id directly
  }
}

// ---------------- K5: segment sums (LDS pre-reduced) ----------------
__global__ void segsum_kernel(const int* __restrict__ assign, const float* __restrict__ each,
                              float* __restrict__ sums, float* __restrict__ cnts) {
  __shared__ float ls[20], lc[20];
  const int t = threadIdx.x;
  if (t < 20) { ls[t] = 0.f; lc[t] = 0.f; }
  __syncthreads();
  const int p = blockIdx.x * 256 + t;
  const int seg = assign[p];
  atomicAdd(&ls[seg], each[p]);
  atomicAdd(&lc[seg], 1.0f);
  __syncthreads();
  if (t < 20) { atomicAdd(&sums[t], ls[t]); atomicAdd(&cnts[t], lc[t]); }
}

// ---------------- K6: segment means + loss_reg ----------------
__global__ void seg_mean_kernel(const float* __restrict__ sums, const float* __restrict__ cnts,
                                float* __restrict__ meanarr, float* __restrict__ loss_reg) {
  __shared__ float red[32];
  const int t = threadIdx.x;
  float reg = 0.0f;
  if (t < 20) {
    const float c = cnts[t], s = sums[t];
    const float mean = s / fmaxf(c, 1.0f);
    meanarr[t] = (c > 0.0f) ? mean : 1.0f;
    reg = (c > 0.0f) ? fmaxf(mean - 0.8f, 0.0f) : 0.0f;
  }
  red[t] = reg;
  __syncthreads();
  for (int o = 16; o > 0; o >>= 1) { if (t < o) red[t] += red[t + o]; __syncthreads(); }
  if (t == 0) loss_reg[0] = red[0] * 0.5f;        // / batch
}

// ---------------- K7: attention map gather, loss_tv, thresholded final ----------------
__global__ void am_tv_kernel(const int* __restrict__ assign, const float* __restrict__ each,
                             const float* __restrict__ meanarr, float* __restrict__ finalarr,
                             float* __restrict__ loss_tv) {
  __shared__ float red[256];
  const int t = threadIdx.x, p = blockIdx.x * 256 + t;
  const float am = meanarr[assign[p]];
  const float d = each[p] - am;
  finalarr[p] = (am < 0.8f) ? 0.0f : am;          // am - stop_grad(am) == 0 numerically
  red[t] = d * d;
  __syncthreads();
  for (int o = 128; o > 0; o >>= 1) { if (t < o) red[t] += red[t + o]; __syncthreads(); }
  if (t == 0) atomicAdd(loss_tv, red[0] * (1.0f / 32768.0f));
}

// ---------------- K8: 5x5 gaussian blur with reflect padding ----------------
__global__ void blur_kernel(const float* __restrict__ finalarr, float* __restrict__ outb) {
  float k1[5]; float ssum = 0.0f;
  #pragma unroll
  for (int j = 0; j < 5; ++j) { float xx = (float)(j - 2) / 1.1f; k1[j] = expf(-0.5f * xx * xx); ssum += k1[j]; }
  #pragma unroll
  for (int j = 0; j < 5; ++j) k1[j] /= ssum;
  const int t = threadIdx.x, p = blockIdx.x * 256 + t;
  const int b = p >> 14, q = p & 16383, y = q >> 7, x = q & 127;
  const float* img = finalarr + b * 16384;
  float acc = 0.0f;
  #pragma unroll
  for (int dy = 0; dy < 5; ++dy) {
    int yy = y + dy - 2; yy = (yy < 0) ? -yy : ((yy > 127) ? 254 - yy : yy);
    const float wy = k1[dy];
    #pragma unroll
    for (int dx = 0; dx < 5; ++dx) {
      int xx = x + dx - 2; xx = (xx < 0) ? -xx : ((xx > 127) ? 254 - xx : xx);
      acc = fmaf(img[yy * 128 + xx], wy * k1[dx], acc);
    }
  }
  outb[p] = acc;
}

// ---------------- launch ----------------
static const int LAYER_NUM_H[17] = {0,2,3,5,6,8,9,11,12,14,15,17,18,20,21,23,24};

extern "C" void kernel_launch(void* const* d_in, const int* in_sizes, int n_in,
                              void* d_out, int out_size, void* d_ws, size_t ws_size,
                              hipStream_t stream) {
  const float* x   = (const float*)d_in[0];
  const float* mw  = (const float*)d_in[27];
  const float* mb  = (const float*)d_in[28];
  const float* afw = (const float*)d_in[29];
  const float* afb = (const float*)d_in[30];
  const float* aw  = (const float*)d_in[31];
  const float* ab  = (const float*)d_in[32];
  const float* alw = (const float*)d_in[33];
  const float* alb = (const float*)d_in[34];
  const float* S   = (const float*)d_in[35];
  const float* blend = (const float*)d_in[1 + 10];   // fmap_10

  FmapPtrs fm;
  fm.p[0] = (const float*)d_in[26];                  // fmap_25
  for (int c = 0; c < 17; ++c) fm.p[1 + c] = (const float*)d_in[1 + LAYER_NUM_H[c]];

  float* out = (float*)d_out;
  float* wsf = (float*)d_ws;
  float* v        = wsf + WS_V;
  float* KC       = wsf + WS_KC;
  float* sums     = wsf + WS_SUMS;
  float* cnts     = wsf + WS_CNTS;
  float* meanarr  = wsf + WS_MEAN;
  float* logitacc = wsf + WS_LOGIT;
  float* finalarr = wsf + WS_FINAL;   // reuses logit area (dead after sigmoid)
  float* each     = wsf + WS_EACH;
  int*   assign   = (int*)(wsf + WS_ASSIGN);

  zero_kernel<<<(WS_ZERO_N + 255) / 256, 256, 0, stream>>>(wsf, WS_ZERO_N);
  zero_kernel<<<(51203 + 255) / 256, 256, 0, stream>>>(out, 51203);

  mapper_kernel<<<14, 256, 0, stream>>>(x, mw, mb, out + OUT_DELTA, out + OUT_LD);
  fold_kernel<<<18, 256, 0, stream>>>(afw, afb, aw, ab, alw, v, KC);
  logits_partial_kernel<<<dim3(NPIX / 256, 44), 256, 0, stream>>>(fm, v, logitacc);
  sigmoid_kernel<<<NPIX / 256, 256, 0, stream>>>(logitacc, KC, alb, each);
  cluster_assign_kernel<<<NPIX / 16 / 8, 256, 0, stream>>>(blend, S, assign);
  segsum_kernel<<<NPIX / 256, 256, 0, stream>>>(assign, each, sums, cnts);
  seg_mean_kernel<<<1, 32, 0, stream>>>(sums, cnts, meanarr, out + OUT_LR);
  am_tv_kernel<<<NPIX / 256, 256, 0, stream>>>(assign, each, meanarr, finalarr, out + OUT_LTV);
  blur_kernel<<<NPIX / 256, 256, 0, stream>>>(finalarr, out + OUT_BLUR);
}